// JPEGEncoder_89524298317929
// MI455X (gfx1250) — compile-verified
//
#include <hip/hip_runtime.h>

// ---------------------------------------------------------------------------
// JPEG encode for MI455X (gfx1250, wave32).
// One wave32 processes a 16x16 pixel tile (4 JPEG 8x8 blocks) using the
// fp32 matrix pipe:  Y16 = D2 * X16 * D2^T  with D2 = diag(D, D), executed as
// 8 chained V_WMMA_F32_16X16X4_F32 ops (two 16x16x16 fp32 matmuls, K=4 each).
// Wave-uniform tile coordinates are forced into SGPRs via readfirstlane so
// all global accesses use scalar-base + 32-bit voffset addressing.
// ---------------------------------------------------------------------------

typedef float v2f __attribute__((ext_vector_type(2)));
typedef float v4f __attribute__((ext_vector_type(4)));
typedef float v8f __attribute__((ext_vector_type(8)));

constexpr int P_   = 8;
constexpr int Bn   = 512;
constexpr int Cn   = 3;
constexpr int Hn   = 128;
constexpr int Wn   = 128;
constexpr int BLKW = Wn / P_;                     // 16 blocks per row
constexpr int NBLK = (Hn / P_) * (Wn / P_);       // 256 blocks per (b,c)
constexpr int TILES_PER_IMG = (Hn / 16) * (Wn / 16);          // 64
constexpr int NTILES = Bn * Cn * TILES_PER_IMG;               // 98304
constexpr int BLKSTRIDE = Bn * Cn * 64;           // 98304 floats per block row
constexpr long long OUT1 = (long long)NBLK * BLKSTRIDE;       // 25165824

#define WAVES_PER_BLOCK 8
#define XS 20  // LDS row stride (floats): 16-byte aligned rows, conflict-free

__global__ __launch_bounds__(256) void jpeg_dct_wmma_kernel(
    const float* __restrict__ img, const float* __restrict__ D,
    const float* __restrict__ Q, float* __restrict__ out) {
  __shared__ float sD2[16 * XS];  // block-diagonal diag(D, D), padded rows
  __shared__ float sQ[64];
  __shared__ float sX[WAVES_PER_BLOCK][16 * XS];
  __shared__ float sT[WAVES_PER_BLOCK][16 * XS];

  const int tid = threadIdx.x;

  // Build D2 in LDS: unconditional load + data-select (v_cndmask, no branch).
  {
    const int m = tid >> 4, kk = tid & 15;
    const float dval = D[(m & 7) * 8 + (kk & 7)];
    sD2[m * XS + kk] = (((m ^ kk) & 8) == 0) ? dval : 0.0f;
    if (tid < 64) sQ[tid] = Q[tid];
  }
  __syncthreads();

  const int wave = tid >> 5;
  const int lane = tid & 31;
  const int l16  = lane & 15;
  const bool hi  = lane >= 16;

  // tileId is wave-uniform: force it into an SGPR so all derived bases are
  // scalar and global accesses use SADDR + 32-bit voffset addressing.
  const int tileId =
      __builtin_amdgcn_readfirstlane(blockIdx.x * WAVES_PER_BLOCK + wave);
  const int tc = tileId & 7;          // tile col within image (8 tiles)
  const int tr = (tileId >> 3) & 7;   // tile row within image
  const int ch = (tileId >> 6) % 3;   // channel
  const int b  = tileId / (3 * TILES_PER_IMG);

  const int row0 = tr * 16, col0 = tc * 16;
  const float* __restrict__ src =
      img + (((size_t)(b * Cn + ch) * Hn + row0) * Wn + col0);  // scalar base

  float* xt = sX[wave];
  float* tt = sT[wave];

  // ---- stage 16x16 tile into LDS (coalesced b128 loads, streamed) ----------
  {
    const int cg = (lane & 3) * 4;  // column group (0,4,8,12)
#pragma unroll
    for (int i = 0; i < 2; ++i) {
      const int r = i * 8 + (lane >> 2);
      const v4f v = __builtin_nontemporal_load(
          reinterpret_cast<const v4f*>(src + r * Wn + cg));
      v4f w;
      w.x = v.x - 128.0f; w.y = v.y - 128.0f;
      w.z = v.z - 128.0f; w.w = v.w - 128.0f;
      *reinterpret_cast<v4f*>(&xt[r * XS + cg]) = w;  // 16B-aligned DS store
    }
  }
  // Same-wave LDS ops are in-order; compiler inserts s_wait_dscnt for RAW.

  // A-operand lane layout (32-bit, 16x4): M = lane&15, K = j + (hi ? 2 : 0)
  // B-operand lane layout (32-bit, 4x16): N = lane&15, K = j + (hi ? 2 : 0)
  // C/D layout (32-bit, 16x16): VGPR v -> M = v + (hi ? 8 : 0), N = lane&15
  //
  // dfrag[k] = D2[l16][4k + j + (hi?2:0)] serves as:
  //   step 1 A fragment:  D2[M][KK]                 (M = l16)
  //   step 2 B fragment:  D2^T[KK][N] = D2[N][KK]   (N = l16)
  v2f dfrag[4];
#pragma unroll
  for (int k = 0; k < 4; ++k) {
#pragma unroll
    for (int j = 0; j < 2; ++j) {
      const int KK = 4 * k + j + (hi ? 2 : 0);
      dfrag[k][j] = sD2[l16 * XS + KK];
    }
  }

  // ---- step 1: T = D2 * X16  (4 x wmma_f32_16x16x4_f32) --------------------
  v8f acc = {0.f, 0.f, 0.f, 0.f, 0.f, 0.f, 0.f, 0.f};
#pragma unroll
  for (int k = 0; k < 4; ++k) {
    v2f bb;
#pragma unroll
    for (int j = 0; j < 2; ++j) {
      const int KK = 4 * k + j + (hi ? 2 : 0);
      bb[j] = xt[KK * XS + l16];
    }
    acc = __builtin_amdgcn_wmma_f32_16x16x4_f32(false, dfrag[k], false, bb,
                                                (short)0, acc, false, false);
  }

  // ---- relayout T through LDS (C/D layout -> A layout) ---------------------
  {
    const int Mb = hi ? 8 : 0;
#pragma unroll
    for (int v = 0; v < 8; ++v) tt[(Mb + v) * XS + l16] = acc[v];
  }

  // ---- step 2: Y = T * D2^T  (4 x wmma_f32_16x16x4_f32) --------------------
  v8f y = {0.f, 0.f, 0.f, 0.f, 0.f, 0.f, 0.f, 0.f};
#pragma unroll
  for (int k = 0; k < 4; ++k) {
    v2f a;
#pragma unroll
    for (int j = 0; j < 2; ++j) {
      const int KK = 4 * k + j + (hi ? 2 : 0);
      a[j] = tt[l16 * XS + KK];  // T[M][KK], M = l16
    }
    y = __builtin_amdgcn_wmma_f32_16x16x4_f32(false, a, false, dfrag[k],
                                              (short)0, y, false, false);
  }

  // ---- quantize + inverse-zigzag scatter -----------------------------------
  // Each lane owns one 8x8 block column: rows r = 0..7 live in y[0..7].
  // Inverse zigzag table packed as 8 bytes per row (byte-select by column).
  constexpr unsigned long long invtab[8] = {
      0x1C1B0F0E06050100ull, 0x2A1D1A100D070402ull, 0x2B291E19110C0803ull,
      0x352C281F18120B09ull, 0x36342D272017130Aull, 0x3C37332E26211614ull,
      0x3D3B38322F252215ull, 0x3F3E3A3931302423ull};

  // Output index = nblk*BLKSTRIDE + b*192 + ch*64 + i, with
  // nblk = (32*tr + 2*tc) [uniform] + (br*16 + bc) [per-lane].
  const int cc = l16 & 7;
  const int nl = (hi ? 16 : 0) + ((l16 >> 3) & 1);    // br*16 + bc
  const int lbase = nl * BLKSTRIDE;                   // 32-bit lane offset
  float* __restrict__ o1u =                           // scalar (SGPR) base
      out + ((size_t)(32 * tr + 2 * tc) * BLKSTRIDE + b * (Cn * 64) + ch * 64);
  float* __restrict__ o2u = o1u + OUT1;

#pragma unroll
  for (int v = 0; v < 8; ++v) {
    const int i   = (int)((invtab[v] >> (cc * 8)) & 63ull);
    const int idx = lbase + i;
    const float dct = y[v];
    const float qv  = rintf(dct / sQ[v * 8 + cc]);    // round-half-even
    __builtin_nontemporal_store(qv, o1u + idx);       // quantized path
    __builtin_nontemporal_store(dct, o2u + idx);      // unquantized path
  }
}

extern "C" void kernel_launch(void* const* d_in, const int* in_sizes, int n_in,
                              void* d_out, int out_size, void* d_ws,
                              size_t ws_size, hipStream_t stream) {
  (void)in_sizes; (void)n_in; (void)d_ws; (void)ws_size; (void)out_size;
  const float* img = (const float*)d_in[0];
  const float* D   = (const float*)d_in[1];
  const float* Q   = (const float*)d_in[2];
  float* out       = (float*)d_out;

  const int nBlocks = NTILES / WAVES_PER_BLOCK;  // 12288
  jpeg_dct_wmma_kernel<<<dim3(nBlocks), dim3(256), 0, stream>>>(img, D, Q, out);
}